// SJLTProjection_37185826848858
// MI455X (gfx1250) — compile-verified
//
#include <hip/hip_runtime.h>

// ---- problem constants (match reference) ----
#define BATCH 64
#define DIM   524288
#define PROJ  8192

// ---- tiling ----
#define BT     4                  // batch rows privatized per block (accumulator = 128 KB)
#define NBT    (BATCH / BT)       // 16 batch tiles
#define NC     32                 // chunks over D
#define DCHUNK (DIM / NC)         // 16384 d's per block
#define TPB    1024               // 32 waves per block
#define DPT    2                  // consecutive d's per thread per iteration
#define DITER  (TPB * DPT)        // 2048 d's per block iteration
#define NITER  (DCHUNK / DITER)   // 8 iterations

#define ACCUM_BYTES   (BT * PROJ * 4)               // 131072 B accumulator
#define STAGE_D_BYTES 32                            // int4 idx + int4 sg per d
#define STAGE_BYTES   (DITER * STAGE_D_BYTES)       // 65536 B per staging buffer
#define SMEM_BYTES    (ACCUM_BYTES + 2 * STAGE_BYTES) // 262144 B total dynamic LDS

static_assert(NITER * DITER == DCHUNK, "");
static_assert(SMEM_BYTES <= 320 * 1024, "");
static_assert((BT * PROJ) % (4 * TPB) == 0, "");

// Zero the 64x8192 fp32 output with b128 stores. Grid covers BATCH*PROJ/4 float4s.
__global__ void sjlt_zero_out(float4* __restrict__ out) {
    int i = blockIdx.x * blockDim.x + threadIdx.x;
    float4 z; z.x = 0.f; z.y = 0.f; z.z = 0.f; z.w = 0.f;
    out[i] = z;
}

// CDNA5 async global->LDS copy (16 B per lane). LDS operand = AS3 byte offset
// (= low 32 bits of the generic shared pointer). Tracked by ASYNCcnt.
__device__ __forceinline__ void async_load_b128_to_lds(unsigned lds_byte_addr, const void* gptr) {
    unsigned long long ga = (unsigned long long)(size_t)gptr;
    asm volatile("global_load_async_to_lds_b128 %0, %1, off"
                 :: "v"(lds_byte_addr), "v"(ga)
                 : "memory");
}

__global__ __launch_bounds__(TPB, 1)
void sjlt_scatter(const float* __restrict__ x,
                  const int4*  __restrict__ idx,   // (DIM,4) int32, rows 16B aligned
                  const int4*  __restrict__ sg,    // (DIM,4) int32 in {0,1}
                  float* __restrict__ out) {
    extern __shared__ unsigned char smem[];
    float* accum = (float*)smem;                   // [BT][PROJ] fp32, bank = col mod 64

    const int nc = blockIdx.x;                     // D chunk
    const int bt = blockIdx.y;                     // batch tile
    const int t  = threadIdx.x;

    const unsigned smem_lds0  = (unsigned)(unsigned long long)(size_t)&smem[0];
    const unsigned stage_lds0 = smem_lds0 + ACCUM_BYTES;
    const unsigned my_slot    = (unsigned)t * (DPT * STAGE_D_BYTES); // this thread's 64 B slot

    const int    d_chunk0 = nc * DCHUNK;
    const float* xbase    = x + (size_t)(bt * BT) * (size_t)DIM;

    // ---- kick off async staging for iteration 0 (each thread stages what it consumes) ----
    {
        const int d = d_chunk0 + t * DPT;
        const unsigned s = stage_lds0 + my_slot;   // buffer 0
        async_load_b128_to_lds(s +  0, &idx[d + 0]);
        async_load_b128_to_lds(s + 16, &sg [d + 0]);
        async_load_b128_to_lds(s + 32, &idx[d + 1]);
        async_load_b128_to_lds(s + 48, &sg [d + 1]);
    }

    // ---- zero the accumulator while staging is in flight (ds_store_b128) ----
    {
        float4 zz; zz.x = 0.f; zz.y = 0.f; zz.z = 0.f; zz.w = 0.f;
        float4* a4 = (float4*)accum;
        #pragma unroll
        for (int i = 0; i < (BT * PROJ / 4) / TPB; ++i)
            a4[i * TPB + t] = zz;
    }
    __syncthreads();

    for (int it = 0; it < NITER; ++it) {
        const int buf = it & 1;
        const int d   = d_chunk0 + it * DITER + t * DPT;

        if (it + 1 < NITER) {
            // stage next iteration into the other buffer, prefetch next x cachelines,
            // then wait for the CURRENT iteration's 4 oldest asyncs (in-order done).
            const int dn = d + DITER;
            const unsigned s = stage_lds0 + (unsigned)((buf ^ 1) * STAGE_BYTES) + my_slot;
            async_load_b128_to_lds(s +  0, &idx[dn + 0]);
            async_load_b128_to_lds(s + 16, &sg [dn + 0]);
            async_load_b128_to_lds(s + 32, &idx[dn + 1]);
            async_load_b128_to_lds(s + 48, &sg [dn + 1]);
            #pragma unroll
            for (int b = 0; b < BT; ++b)
                __builtin_prefetch(&xbase[(size_t)b * DIM + dn], 0, 1);
            asm volatile("s_wait_asynccnt 0x4" ::: "memory");
        } else {
            asm volatile("s_wait_asynccnt 0x0" ::: "memory");
        }

        // read this thread's staged idx/sign rows back from LDS (ds_load_b128)
        const int4* st = (const int4*)(smem + ACCUM_BYTES + buf * STAGE_BYTES
                                       + t * (DPT * STAGE_D_BYTES));
        const int4 i0 = st[0];
        const int4 s0 = st[1];
        const int4 i1 = st[2];
        const int4 s1 = st[3];

        // fp32 sign-flip masks: sg==1 -> +x, sg==0 -> -x (flip sign bit)
        const unsigned m0[4] = { (unsigned)(s0.x ^ 1) << 31, (unsigned)(s0.y ^ 1) << 31,
                                 (unsigned)(s0.z ^ 1) << 31, (unsigned)(s0.w ^ 1) << 31 };
        const unsigned m1[4] = { (unsigned)(s1.x ^ 1) << 31, (unsigned)(s1.y ^ 1) << 31,
                                 (unsigned)(s1.z ^ 1) << 31, (unsigned)(s1.w ^ 1) << 31 };

        #pragma unroll
        for (int b = 0; b < BT; ++b) {
            // coalesced 8B load of x[b, d..d+1] (wave covers 256 contiguous bytes)
            const float2 xv = *(const float2*)&xbase[(size_t)b * DIM + d];
            const unsigned x0 = __float_as_uint(xv.x);
            const unsigned x1 = __float_as_uint(xv.y);
            float* row = accum + b * PROJ;
            // 8 scatter-adds -> ds_add_f32 (no-return LDS float atomics),
            // addresses uniformly random over all 64 banks.
            atomicAdd(&row[i0.x], __uint_as_float(x0 ^ m0[0]));
            atomicAdd(&row[i0.y], __uint_as_float(x0 ^ m0[1]));
            atomicAdd(&row[i0.z], __uint_as_float(x0 ^ m0[2]));
            atomicAdd(&row[i0.w], __uint_as_float(x0 ^ m0[3]));
            atomicAdd(&row[i1.x], __uint_as_float(x1 ^ m1[0]));
            atomicAdd(&row[i1.y], __uint_as_float(x1 ^ m1[1]));
            atomicAdd(&row[i1.z], __uint_as_float(x1 ^ m1[2]));
            atomicAdd(&row[i1.w], __uint_as_float(x1 ^ m1[3]));
        }
    }
    __syncthreads();

    // ---- flush: scale by 1/sqrt(C)=0.5 and reduce across D-chunks via fp32 global atomics ----
    float* obase = out + (size_t)(bt * BT) * (size_t)PROJ;
    for (int i = 0; i < (BT * PROJ) / TPB; ++i) {
        const int off = i * TPB + t;                       // coalesced across the block
        unsafeAtomicAdd(&obase[off], 0.5f * accum[off]);   // -> global_atomic_add_f32
    }
}

extern "C" void kernel_launch(void* const* d_in, const int* in_sizes, int n_in,
                              void* d_out, int out_size, void* d_ws, size_t ws_size,
                              hipStream_t stream) {
    (void)in_sizes; (void)n_in; (void)out_size; (void)d_ws; (void)ws_size;

    const float* x   = (const float*)d_in[0];
    const int4*  idx = (const int4*)d_in[1];
    const int4*  sg  = (const int4*)d_in[2];
    float*       out = (float*)d_out;

    // 1) zero the output (harness poisons it; atomics below accumulate from zero)
    {
        const int nvec = (BATCH * PROJ) / 4;    // 131072 float4s
        sjlt_zero_out<<<nvec / 256, 256, 0, stream>>>((float4*)out);
    }

    // 2) LDS-privatized scatter-add with async-staged index/sign stream
    {
        dim3 grid(NC, NBT);                     // 32 x 16 = 512 blocks
        dim3 block(TPB);                        // 1024 threads = 32 waves
        sjlt_scatter<<<grid, block, SMEM_BYTES, stream>>>(x, idx, sg, out);
    }
}